// MemoryBank_62173946577471
// MI455X (gfx1250) — compile-verified
//
#include <hip/hip_runtime.h>

typedef __attribute__((ext_vector_type(2))) float v2f;
typedef __attribute__((ext_vector_type(4))) float v4f;
typedef __attribute__((ext_vector_type(8))) float v8f;

namespace {
constexpr int   kB    = 4096;   // batch
constexpr int   kS    = 5;      // slots per class
constexpr int   kD    = 2048;   // feature dim
constexpr int   kK    = 3;      // top_k (min(3, S))
constexpr float kEPS  = 1e-8f;
constexpr float kTEMP = 0.1f;
}

// Read a float from a fixed lane (wave-uniform, constant lane after unroll).
static __device__ __forceinline__ float rdlane(float x, int l) {
  return __builtin_bit_cast(
      float, __builtin_amdgcn_readlane(__builtin_bit_cast(int, x), l));
}

// One wave (32 threads) handles TWO samples.
// Phase 1: Gram matrix of X = [q, k0..k4] (sample0 rows 0..5, sample1 rows 8..13)
//          via v_wmma_f32_16x16x4_f32 with A == B (per-lane layouts coincide),
//          using 4 independent accumulators to break the WMMA RAW chain.
// Phase 2: wave-uniform top-k/softmax via v_readlane, then coalesced float4
//          blend of the 3 chosen value rows.
__global__ __launch_bounds__(32) void memorybank_wmma_kernel(
    const float* __restrict__ query,    // (B, D)
    const int*   __restrict__ labels,   // (B,)
    const float* __restrict__ mkeys,    // (C, S, D)
    const float* __restrict__ mvals,    // (C, S, D)
    const float* __restrict__ qual,     // (C, S)
    float* __restrict__ outR,           // (B, D)
    float* __restrict__ outW)           // (B,)
{
  const int lane = threadIdx.x & 31;
  const int b0   = blockIdx.x * 2;      // samples b0, b0+1
  const int h    = lane >> 4;           // K-half: 0 -> K{0,1}, 1 -> K{2,3}
  const int mi   = lane & 15;           // matrix row index M
  const int samp = mi >> 3;             // which sample this row belongs to
  const int r    = mi & 7;              // 0 = query, 1..5 = key slot, 6/7 dummy
  const int bl   = b0 + samp;

  // Row pointer for this lane's matrix row. Rows 6/7 duplicate the query row:
  // their Gram entries land only in C rows/cols {6,7,14,15}, which are never read.
  const int lab_lane = labels[bl];
  const float* rowp;
  if (r >= 1 && r <= 5) {
    rowp = mkeys + ((size_t)lab_lane * kS + (r - 1)) * (size_t)kD;
  } else {
    rowp = query + (size_t)bl * (size_t)kD;
  }
  rowp += 2 * h;

  // ---- Phase 1: f32 WMMA Gram accumulate over D, 4 independent chains ----
  v8f c0 = {}, c1 = {}, c2 = {}, c3 = {};
#pragma unroll 2
  for (int k = 0; k < kD; k += 16) {
    const v2f a0 = *(const v2f*)(rowp + k);
    const v2f a1 = *(const v2f*)(rowp + k + 4);
    const v2f a2 = *(const v2f*)(rowp + k + 8);
    const v2f a3 = *(const v2f*)(rowp + k + 12);
    c0 = __builtin_amdgcn_wmma_f32_16x16x4_f32(false, a0, false, a0,
                                               (short)0, c0, false, false);
    c1 = __builtin_amdgcn_wmma_f32_16x16x4_f32(false, a1, false, a1,
                                               (short)0, c1, false, false);
    c2 = __builtin_amdgcn_wmma_f32_16x16x4_f32(false, a2, false, a2,
                                               (short)0, c2, false, false);
    c3 = __builtin_amdgcn_wmma_f32_16x16x4_f32(false, a3, false, a3,
                                               (short)0, c3, false, false);
  }
  const v8f c = (c0 + c1) + (c2 + c3);

  // ---- Phase 2: per-sample scalar math + value blend ----
#pragma unroll
  for (int ss = 0; ss < 2; ++ss) {
    const int b    = b0 + ss;
    const int lab  = labels[b];
    const int base = ss * 24;  // sample0: lanes 0..5 ; sample1: lanes 24..29

    // C[m][n]: m<8 -> c[m] @ lane n ; m>=8 -> c[m-8] @ lane 16+n
    const float qq = rdlane(c[0], base);
    const float qn = fmaxf(sqrtf(qq), kEPS);

    float comb[kS];
    float ssum = 0.f;
#pragma unroll
    for (int s = 0; s < kS; ++s) {
      const float qk = rdlane(c[0],     base + 1 + s);  // C[row_q][row_ks]
      const float kk = rdlane(c[s + 1], base + 1 + s);  // C[row_ks][row_ks]
      const float kn = fmaxf(sqrtf(kk), kEPS);
      const float sc = qual[(size_t)lab * kS + s];
      ssum   += sc;
      comb[s] = (qk / (qn * kn)) * sc;
    }
    const bool hit = (ssum != 0.f);

    // top-3 of 5, strict '>' keeps lowest index on ties (matches lax.top_k)
    float w0 = comb[0], w1 = comb[1], w2 = comb[2], w3 = comb[3], w4 = comb[4];
    float topv[kK]; int topi[kK];
#pragma unroll
    for (int j = 0; j < kK; ++j) {
      float m = w0; int bi = 0;
      if (w1 > m) { m = w1; bi = 1; }
      if (w2 > m) { m = w2; bi = 2; }
      if (w3 > m) { m = w3; bi = 3; }
      if (w4 > m) { m = w4; bi = 4; }
      topv[j] = m; topi[j] = bi;
      if      (bi == 0) w0 = -__builtin_inff();
      else if (bi == 1) w1 = -__builtin_inff();
      else if (bi == 2) w2 = -__builtin_inff();
      else if (bi == 3) w3 = -__builtin_inff();
      else              w4 = -__builtin_inff();
    }

    // softmax(top/T), stable (topv[0] is the max)
    const float invT = 1.0f / kTEMP;
    const float e0 = 1.0f;  // exp(0)
    const float e1 = __expf((topv[1] - topv[0]) * invT);
    const float e2 = __expf((topv[2] - topv[0]) * invT);
    const float inv = 1.0f / (e0 + e1 + e2);
    const float a0 = hit ? e0 * inv : 0.f;
    const float a1 = hit ? e1 * inv : 0.f;
    const float a2 = hit ? e2 * inv : 0.f;
    const float wt = hit ? (topv[0] + topv[1] + topv[2]) * (1.0f / 3.0f) : 0.f;

    const float* v0 = mvals + ((size_t)lab * kS + topi[0]) * (size_t)kD;
    const float* v1 = mvals + ((size_t)lab * kS + topi[1]) * (size_t)kD;
    const float* v2 = mvals + ((size_t)lab * kS + topi[2]) * (size_t)kD;
    float* orow = outR + (size_t)b * (size_t)kD;

    // Coalesced: lane i covers 16B, wave covers 512B per iteration.
#pragma unroll 4
    for (int d = lane * 4; d < kD; d += 32 * 4) {
      const v4f x0 = *(const v4f*)(v0 + d);
      const v4f x1 = *(const v4f*)(v1 + d);
      const v4f x2 = *(const v4f*)(v2 + d);
      v4f res = x0 * a0 + x1 * a1 + x2 * a2;
      *(v4f*)(orow + d) = res;
    }
    if (lane == 0) outW[b] = wt;
  }
}

extern "C" void kernel_launch(void* const* d_in, const int* in_sizes, int n_in,
                              void* d_out, int out_size, void* d_ws, size_t ws_size,
                              hipStream_t stream) {
  const float* q   = (const float*)d_in[0];  // query        (B, D) f32
  const int*   lab = (const int*)  d_in[1];  // labels       (B,)   i32
  const float* mk  = (const float*)d_in[2];  // memory_keys  (C,S,D) f32
  const float* mv  = (const float*)d_in[3];  // memory_values(C,S,D) f32
  const float* qs  = (const float*)d_in[4];  // quality      (C,S)  f32
  // d_in[5] = top_k (==3), baked in as kK.

  float* outR = (float*)d_out;
  float* outW = outR + (size_t)kB * (size_t)kD;

  dim3 grid(kB / 2);
  dim3 block(32);
  hipLaunchKernelGGL(memorybank_wmma_kernel, grid, block, 0, stream,
                     q, lab, mk, mv, qs, outR, outW);
}